// SynergySAGE_48155173322905
// MI455X (gfx1250) — compile-verified
//
#include <hip/hip_runtime.h>
#include <hip/hip_bf16.h>

#define N_NODES   50000
#define N_EDGES   800000
#define NUM_GRAPHS 256
#define HID       128
#define IN_DIM    64
#define BN_EPS    1e-5f
#define LDSS      132   // padded LDS row stride (floats): 132 % 64 = 4 -> conflict-free frag reads

typedef float v2f __attribute__((ext_vector_type(2)));
typedef float v8f __attribute__((ext_vector_type(8)));

// ---------------- degree ----------------
__global__ __launch_bounds__(256) void deg_kernel(const long long* __restrict__ dst,
                                                  float* __restrict__ deg) {
    int e = blockIdx.x * 256 + threadIdx.x;
    if (e < N_EDGES) atomicAdd(&deg[(int)dst[e]], 1.0f);
}

__global__ __launch_bounds__(256) void invdeg_kernel(float* __restrict__ deg) {
    int i = blockIdx.x * 256 + threadIdx.x;
    if (i < N_NODES) deg[i] = 1.0f / fmaxf(deg[i], 1.0f);
}

// ---------------- edge aggregation: wave per edge, lanes stride channels ----------------
__global__ __launch_bounds__(256) void agg_kernel(const float* __restrict__ hin,
                                                  const long long* __restrict__ src,
                                                  const long long* __restrict__ dst,
                                                  float* __restrict__ agg, int C) {
    int wave = (blockIdx.x * 256 + threadIdx.x) >> 5;
    int lane = threadIdx.x & 31;
    if (wave >= N_EDGES) return;
    int s = (int)src[wave];
    int d = (int)dst[wave];
    const float* hs = hin + (long)s * C;
    float* ad = agg + (long)d * C;
    for (int c = lane; c < C; c += 32) atomicAdd(&ad[c], hs[c]);
}

// ---------------- SAGE GEMM: H = (agg*inv_deg) @ Wn + root @ Wr + b  (fp32 WMMA) ----------
// block = 256 threads = 8 waves; block owns rows [m0, m0+16); wave w owns cols [16w, 16w+16)
__global__ __launch_bounds__(256) void sage_gemm_kernel(
        const float* __restrict__ Aagg, const float* __restrict__ Aroot,
        const float* __restrict__ invdeg,
        const float* __restrict__ Wn, const float* __restrict__ Wr,
        const float* __restrict__ bias, float* __restrict__ Hout,
        int K, int kshift) {
    __shared__ float s_agg [16 * LDSS];
    __shared__ float s_root[16 * LDSS];

    const int tid = threadIdx.x;
    const int m0  = blockIdx.x * 16;

    // cooperative, coalesced stage of both A tiles (scale agg rows by inv_deg here)
    for (int idx = tid; idx < 16 * K; idx += 256) {
        int r = idx >> kshift;
        int c = idx & (K - 1);
        float id = invdeg[m0 + r];
        s_agg [r * LDSS + c] = Aagg [(long)(m0 + r) * K + c] * id;
        s_root[r * LDSS + c] = Aroot[(long)(m0 + r) * K + c];
    }
    __syncthreads();

    const int w    = tid >> 5;
    const int lane = tid & 31;
    const int half = lane >> 4;   // 0: K={0,1}, 1: K={2,3} of each 4-step
    const int lrow = lane & 15;
    const int n0   = w * 16;

    const float* wn = Wn + n0 + lrow;   // column n0+lrow of Wn (K x 128 row-major)
    const float* wr = Wr + n0 + lrow;

    v8f acc = {};
    for (int kk = 0; kk < K; kk += 4) {
        const int ka = kk + 2 * half;
        v2f a, b;
        // neighbor-aggregate term
        a.x = s_agg[lrow * LDSS + ka];
        a.y = s_agg[lrow * LDSS + ka + 1];
        b.x = wn[(long)ka * HID];
        b.y = wn[(long)(ka + 1) * HID];
        acc = __builtin_amdgcn_wmma_f32_16x16x4_f32(false, a, false, b,
                                                    (short)0, acc, false, false);
        // root term
        a.x = s_root[lrow * LDSS + ka];
        a.y = s_root[lrow * LDSS + ka + 1];
        b.x = wr[(long)ka * HID];
        b.y = wr[(long)(ka + 1) * HID];
        acc = __builtin_amdgcn_wmma_f32_16x16x4_f32(false, a, false, b,
                                                    (short)0, acc, false, false);
    }

    const float bv = bias[n0 + lrow];
    #pragma unroll
    for (int v = 0; v < 8; ++v) {
        int row = m0 + v + 8 * half;           // C/D layout: VGPR v -> M = v + 8*half
        Hout[(long)row * HID + n0 + lrow] = acc[v] + bv;
    }
}

// ---------------- batch-norm stats: per-channel sum / sumsq ----------------
__global__ __launch_bounds__(128) void bn_stats_kernel(const float* __restrict__ H,
                                                       float* __restrict__ gsum,
                                                       float* __restrict__ gsq,
                                                       int rowsPerBlock) {
    int c  = threadIdx.x;
    int r0 = blockIdx.x * rowsPerBlock;
    int r1 = min(r0 + rowsPerBlock, N_NODES);
    float s = 0.0f, q = 0.0f;
    for (int r = r0; r < r1; ++r) {
        float v = H[(long)r * HID + c];
        s += v; q += v * v;
    }
    atomicAdd(&gsum[c], s);
    atomicAdd(&gsq[c],  q);
}

// ---------------- batch-norm apply + ReLU ----------------
__global__ __launch_bounds__(256) void bn_apply_kernel(const float* __restrict__ Hin,
                                                       const float* __restrict__ gsum,
                                                       const float* __restrict__ gsq,
                                                       const float* __restrict__ gamma,
                                                       const float* __restrict__ beta,
                                                       float* __restrict__ Hout) {
    long i = (long)blockIdx.x * 256 + threadIdx.x;
    if (i >= (long)N_NODES * HID) return;
    int c = (int)(i & (HID - 1));
    const float invN = 1.0f / (float)N_NODES;
    float mu  = gsum[c] * invN;
    float var = gsq[c] * invN - mu * mu;
    float sc  = gamma[c] * rsqrtf(var + BN_EPS);
    float v   = (Hin[i] - mu) * sc + beta[c];
    Hout[i] = fmaxf(v, 0.0f);
}

// ---------------- global mean pool (sum + count), wave per node ----------------
__global__ __launch_bounds__(256) void pool_kernel(const float* __restrict__ H,
                                                   const long long* __restrict__ batch,
                                                   float* __restrict__ pooled,
                                                   float* __restrict__ cnt) {
    int wave = (blockIdx.x * 256 + threadIdx.x) >> 5;
    int lane = threadIdx.x & 31;
    if (wave >= N_NODES) return;
    int g = (int)batch[wave];
    const float* hr = H + (long)wave * HID;
    float* pg = pooled + g * HID;
    for (int c = lane; c < HID; c += 32) atomicAdd(&pg[c], hr[c]);
    if (lane == 0) atomicAdd(&cnt[g], 1.0f);
}

// ---------------- head MLP: 256 threads = 256 graphs ----------------
__global__ __launch_bounds__(256) void mlp_kernel(const float* __restrict__ pooled,
                                                  const float* __restrict__ cnt,
                                                  const float* __restrict__ fc1W,
                                                  const float* __restrict__ fc1b,
                                                  const float* __restrict__ fc2W,
                                                  const float* __restrict__ fc2b,
                                                  float* __restrict__ out) {
    int g = threadIdx.x;
    float inv = 1.0f / fmaxf(cnt[g], 1.0f);
    const float* pg = pooled + g * HID;
    float acc = fc2b[0];
    for (int j = 0; j < HID / 2; ++j) {
        float z = fc1b[j];
        for (int c = 0; c < HID; ++c)
            z += pg[c] * inv * fc1W[c * (HID / 2) + j];
        acc += fmaxf(z, 0.0f) * fc2W[j];
    }
    out[g] = 1.0f / (1.0f + expf(-acc));
}

extern "C" void kernel_launch(void* const* d_in, const int* in_sizes, int n_in,
                              void* d_out, int out_size, void* d_ws, size_t ws_size,
                              hipStream_t stream) {
    const float*     x     = (const float*)d_in[0];
    const long long* eidx  = (const long long*)d_in[1];
    const long long* batch = (const long long*)d_in[2];
    const float* Wn0 = (const float*)d_in[3];
    const float* Wr0 = (const float*)d_in[4];
    const float* b0  = (const float*)d_in[5];
    const float* Wn1 = (const float*)d_in[6];
    const float* Wr1 = (const float*)d_in[7];
    const float* b1  = (const float*)d_in[8];
    const float* Wn2 = (const float*)d_in[9];
    const float* Wr2 = (const float*)d_in[10];
    const float* b2  = (const float*)d_in[11];
    const float* gamma = (const float*)d_in[12];   // [3,128]
    const float* beta  = (const float*)d_in[13];   // [3,128]
    const float* fc1W  = (const float*)d_in[14];
    const float* fc1b  = (const float*)d_in[15];
    const float* fc2W  = (const float*)d_in[16];
    const float* fc2b  = (const float*)d_in[17];
    float* out = (float*)d_out;

    const long long* src = eidx;
    const long long* dst = eidx + N_EDGES;

    // workspace carve-out
    char* ws = (char*)d_ws;
    size_t off = 0;
    auto carve = [&](size_t bytes) -> void* {
        void* p = ws + off;
        off += (bytes + 255) & ~(size_t)255;
        return p;
    };
    const size_t featBytes = (size_t)N_NODES * HID * sizeof(float);
    float* deg    = (float*)carve((size_t)N_NODES * sizeof(float));
    float* agg    = (float*)carve(featBytes);
    float* P      = (float*)carve(featBytes);   // pre-BN
    float* Q      = (float*)carve(featBytes);   // post-BN / layer input
    float* gsum   = (float*)carve(HID * sizeof(float));
    float* gsq    = (float*)carve(HID * sizeof(float));
    float* pooled = (float*)carve((size_t)NUM_GRAPHS * HID * sizeof(float));
    float* cnt    = (float*)carve((size_t)NUM_GRAPHS * sizeof(float));

    const int edgeBlocks = N_EDGES / 8;      // wave per edge, 8 waves/block
    const int tileBlocks = N_NODES / 16;     // 3125 exact
    const int bnRows = 256;
    const int bnBlocks = (N_NODES + bnRows - 1) / bnRows;
    const int ewBlocks = (int)(((long)N_NODES * HID + 255) / 256);

    // degrees
    hipMemsetAsync(deg, 0, (size_t)N_NODES * sizeof(float), stream);
    deg_kernel<<<(N_EDGES + 255) / 256, 256, 0, stream>>>(dst, deg);
    invdeg_kernel<<<(N_NODES + 255) / 256, 256, 0, stream>>>(deg);

    // ---- layer 0 (64 -> 128) ----
    hipMemsetAsync(agg, 0, (size_t)N_NODES * IN_DIM * sizeof(float), stream);
    agg_kernel<<<edgeBlocks, 256, 0, stream>>>(x, src, dst, agg, IN_DIM);
    sage_gemm_kernel<<<tileBlocks, 256, 0, stream>>>(agg, x, deg, Wn0, Wr0, b0, P, IN_DIM, 6);
    hipMemsetAsync(gsum, 0, HID * sizeof(float), stream);
    hipMemsetAsync(gsq,  0, HID * sizeof(float), stream);
    bn_stats_kernel<<<bnBlocks, 128, 0, stream>>>(P, gsum, gsq, bnRows);
    bn_apply_kernel<<<ewBlocks, 256, 0, stream>>>(P, gsum, gsq, gamma + 0 * HID, beta + 0 * HID, Q);

    // ---- layer 1 (128 -> 128) ----
    hipMemsetAsync(agg, 0, featBytes, stream);
    agg_kernel<<<edgeBlocks, 256, 0, stream>>>(Q, src, dst, agg, HID);
    sage_gemm_kernel<<<tileBlocks, 256, 0, stream>>>(agg, Q, deg, Wn1, Wr1, b1, P, HID, 7);
    hipMemsetAsync(gsum, 0, HID * sizeof(float), stream);
    hipMemsetAsync(gsq,  0, HID * sizeof(float), stream);
    bn_stats_kernel<<<bnBlocks, 128, 0, stream>>>(P, gsum, gsq, bnRows);
    bn_apply_kernel<<<ewBlocks, 256, 0, stream>>>(P, gsum, gsq, gamma + 1 * HID, beta + 1 * HID, Q);

    // ---- layer 2 (128 -> 128) ----
    hipMemsetAsync(agg, 0, featBytes, stream);
    agg_kernel<<<edgeBlocks, 256, 0, stream>>>(Q, src, dst, agg, HID);
    sage_gemm_kernel<<<tileBlocks, 256, 0, stream>>>(agg, Q, deg, Wn2, Wr2, b2, P, HID, 7);
    hipMemsetAsync(gsum, 0, HID * sizeof(float), stream);
    hipMemsetAsync(gsq,  0, HID * sizeof(float), stream);
    bn_stats_kernel<<<bnBlocks, 128, 0, stream>>>(P, gsum, gsq, bnRows);
    bn_apply_kernel<<<ewBlocks, 256, 0, stream>>>(P, gsum, gsq, gamma + 2 * HID, beta + 2 * HID, Q);

    // ---- pool + MLP head ----
    hipMemsetAsync(pooled, 0, (size_t)NUM_GRAPHS * HID * sizeof(float), stream);
    hipMemsetAsync(cnt,    0, (size_t)NUM_GRAPHS * sizeof(float), stream);
    pool_kernel<<<N_NODES / 8, 256, 0, stream>>>(Q, batch, pooled, cnt);
    mlp_kernel<<<1, 256, 0, stream>>>(pooled, cnt, fc1W, fc1b, fc2W, fc2b, out);
}